// GINConv_16604343566552
// MI455X (gfx1250) — compile-verified
//
#include <hip/hip_runtime.h>
#include <hip/hip_bf16.h>

typedef float v2f __attribute__((ext_vector_type(2)));
typedef float v8f __attribute__((ext_vector_type(8)));

#define N_NODES 50000
#define N_EDGES 800000
#define D_IN    128
#define D_HID   256
#define D_OUT   128
#define BN_EPS  1e-5f

#define LDA1 132   // 16x128 A tile pad: 132 % 64 == 4 -> conflict-free column reads
#define LDA2 260   // 16x256 A tile pad: 260 % 64 == 4

// ---------------------------------------------------------------- init / copy
__global__ __launch_bounds__(256) void k_copy_f4(const float* __restrict__ src,
                                                 float* __restrict__ dst, int n4) {
    int i = blockIdx.x * blockDim.x + threadIdx.x;
    if (i < n4) ((float4*)dst)[i] = ((const float4*)src)[i];
}

__global__ __launch_bounds__(512) void k_zero_stats(float* __restrict__ stats) {
    stats[threadIdx.x] = 0.0f;   // 512 floats: colsum[256] ++ colsumsq[256]
}

// ---------------------------------------------------------------- edge scatter
// One wave (32 lanes) per edge; lane handles 4 feature dims.
__global__ __launch_bounds__(256) void k_scatter(const float* __restrict__ x,
                                                 const long long* __restrict__ ei,
                                                 float* __restrict__ agg) {
    int gtid = blockIdx.x * blockDim.x + threadIdx.x;
    int e    = gtid >> 5;
    int lane = gtid & 31;
    if (e >= N_EDGES) return;
    long long s = ei[e];
    long long d = ei[N_EDGES + e];
    float4 v = *(const float4*)(x + s * D_IN + lane * 4);
    float* p = agg + d * D_IN + lane * 4;
    atomicAdd(p + 0, v.x);
    atomicAdd(p + 1, v.y);
    atomicAdd(p + 2, v.z);
    atomicAdd(p + 3, v.w);
}

// ---------------------------------------------------------------- GEMM1 + BN stats
// h1[16m:16m+16, 0:256] = agg_tile @ W1 + b1 ; accumulate col sums / sumsq.
// Block = 128 threads = 4 waves. Wave w owns columns [64w, 64w+64) as 4 WMMA tiles.
__global__ __launch_bounds__(128) void k_gemm1(const float* __restrict__ agg,
                                               const float* __restrict__ W1,
                                               const float* __restrict__ b1,
                                               float* __restrict__ h1,
                                               float* __restrict__ colsum,
                                               float* __restrict__ colsumsq) {
    __shared__ float As[16 * LDA1];
    const int m   = blockIdx.x;
    const int tid = threadIdx.x;

    // Stage A tile (16 x 128 fp32) coalesced: thread t -> row t/8, 16 floats.
    {
        int row  = tid >> 3;
        int colb = (tid & 7) * 16;
        const float* s = agg + (m * 16 + row) * D_IN + colb;
        float*       d = As + row * LDA1 + colb;
#pragma unroll
        for (int i = 0; i < 4; ++i)
            *(float4*)(d + i * 4) = *(const float4*)(s + i * 4);
    }
    __syncthreads();

    const int wave    = tid >> 5;
    const int lane    = tid & 31;
    const int half    = lane >> 4;   // 0: K offset 0, 1: K offset 2
    const int lane16  = lane & 15;
    const int ncb     = wave * 64;

    v8f c[4] = {};
    for (int k = 0; k < D_IN; k += 4) {
        const int kk = k + 2 * half;
        v2f a;
        a.x = As[lane16 * LDA1 + kk];
        a.y = As[lane16 * LDA1 + kk + 1];
#pragma unroll
        for (int t = 0; t < 4; ++t) {
            const int col = ncb + t * 16 + lane16;
            v2f b;
            b.x = W1[kk * D_HID + col];
            b.y = W1[(kk + 1) * D_HID + col];
            c[t] = __builtin_amdgcn_wmma_f32_16x16x4_f32(
                false, a, false, b, (short)0, c[t], false, false);
        }
    }

    // Epilogue: +bias, store h1, accumulate BN statistics.
#pragma unroll
    for (int t = 0; t < 4; ++t) {
        const int col  = ncb + t * 16 + lane16;
        const float bs = b1[col];
        float s = 0.0f, sq = 0.0f;
        float* out = h1 + (m * 16 + 8 * half) * D_HID + col;
#pragma unroll
        for (int r = 0; r < 8; ++r) {
            float v = c[t][r] + bs;
            out[r * D_HID] = v;
            s  += v;
            sq += v * v;
        }
        atomicAdd(&colsum[col],   s);
        atomicAdd(&colsumsq[col], sq);
    }
}

// ---------------------------------------------------------------- BN finalize
__global__ __launch_bounds__(256) void k_bn_finalize(const float* __restrict__ colsum,
                                                     const float* __restrict__ colsumsq,
                                                     const float* __restrict__ gamma,
                                                     const float* __restrict__ beta,
                                                     float* __restrict__ scale,
                                                     float* __restrict__ shift) {
    int c = threadIdx.x;
    const float inv_n = 1.0f / (float)N_NODES;
    float mean = colsum[c] * inv_n;
    float var  = colsumsq[c] * inv_n - mean * mean;
    float sc   = gamma[c] * rsqrtf(var + BN_EPS);
    scale[c] = sc;
    shift[c] = beta[c] - mean * sc;
}

// ---------------------------------------------------------------- GEMM2
// out = relu(h1*scale+shift) @ W2 + b2. BN+ReLU fused into the A-tile stage.
__global__ __launch_bounds__(128) void k_gemm2(const float* __restrict__ h1,
                                               const float* __restrict__ scale,
                                               const float* __restrict__ shift,
                                               const float* __restrict__ W2,
                                               const float* __restrict__ b2,
                                               float* __restrict__ out) {
    __shared__ float As[16 * LDA2];
    __shared__ float sSc[D_HID];
    __shared__ float sSh[D_HID];
    const int m   = blockIdx.x;
    const int tid = threadIdx.x;

    sSc[tid * 2]     = scale[tid * 2];
    sSc[tid * 2 + 1] = scale[tid * 2 + 1];
    sSh[tid * 2]     = shift[tid * 2];
    sSh[tid * 2 + 1] = shift[tid * 2 + 1];
    __syncthreads();

    // Stage A tile (16 x 256) with BN + ReLU applied: thread t -> row t/8, 32 floats.
    {
        int row  = tid >> 3;
        int colb = (tid & 7) * 32;
        const float* s = h1 + (m * 16 + row) * D_HID + colb;
        float*       d = As + row * LDA2 + colb;
#pragma unroll
        for (int i = 0; i < 8; ++i) {
            float4 v = *(const float4*)(s + i * 4);
            int c = colb + i * 4;
            v.x = fmaxf(v.x * sSc[c + 0] + sSh[c + 0], 0.0f);
            v.y = fmaxf(v.y * sSc[c + 1] + sSh[c + 1], 0.0f);
            v.z = fmaxf(v.z * sSc[c + 2] + sSh[c + 2], 0.0f);
            v.w = fmaxf(v.w * sSc[c + 3] + sSh[c + 3], 0.0f);
            *(float4*)(d + i * 4) = v;
        }
    }
    __syncthreads();

    const int wave   = tid >> 5;
    const int lane   = tid & 31;
    const int half   = lane >> 4;
    const int lane16 = lane & 15;
    const int ncb    = wave * 32;   // 4 waves x 32 cols = 128 output cols

    v8f c[2] = {};
    for (int k = 0; k < D_HID; k += 4) {
        const int kk = k + 2 * half;
        v2f a;
        a.x = As[lane16 * LDA2 + kk];
        a.y = As[lane16 * LDA2 + kk + 1];
#pragma unroll
        for (int t = 0; t < 2; ++t) {
            const int col = ncb + t * 16 + lane16;
            v2f b;
            b.x = W2[kk * D_OUT + col];
            b.y = W2[(kk + 1) * D_OUT + col];
            c[t] = __builtin_amdgcn_wmma_f32_16x16x4_f32(
                false, a, false, b, (short)0, c[t], false, false);
        }
    }

#pragma unroll
    for (int t = 0; t < 2; ++t) {
        const int col  = ncb + t * 16 + lane16;
        const float bs = b2[col];
        float* o = out + (m * 16 + 8 * half) * D_OUT + col;
#pragma unroll
        for (int r = 0; r < 8; ++r)
            o[r * D_OUT] = c[t][r] + bs;
    }
}

// ---------------------------------------------------------------- launch
extern "C" void kernel_launch(void* const* d_in, const int* in_sizes, int n_in,
                              void* d_out, int out_size, void* d_ws, size_t ws_size,
                              hipStream_t stream) {
    const float*     x     = (const float*)d_in[0];
    const long long* ei    = (const long long*)d_in[1];   // int64 edge index
    const float*     W1    = (const float*)d_in[2];
    const float*     b1    = (const float*)d_in[3];
    const float*     gamma = (const float*)d_in[4];
    const float*     beta  = (const float*)d_in[5];
    const float*     W2    = (const float*)d_in[6];
    const float*     b2    = (const float*)d_in[7];
    float* out = (float*)d_out;

    // Workspace layout
    char* ws = (char*)d_ws;
    float* agg      = (float*)(ws);                                        // N*128
    float* h1       = (float*)(ws + (size_t)N_NODES * D_IN  * 4);          // N*256
    float* stats    = (float*)(ws + (size_t)N_NODES * (D_IN + D_HID) * 4); // 512 + 512
    float* colsum   = stats;
    float* colsumsq = stats + 256;
    float* scale    = stats + 512;
    float* shift    = stats + 768;

    // 1) agg = x ; zero BN stat accumulators
    {
        int n4 = N_NODES * D_IN / 4;
        k_copy_f4<<<(n4 + 255) / 256, 256, 0, stream>>>(x, agg, n4);
        k_zero_stats<<<1, 512, 0, stream>>>(colsum);
    }
    // 2) scatter-sum over edges (1 wave per edge)
    k_scatter<<<N_EDGES / 8, 256, 0, stream>>>(x, ei, agg);
    // 3) GEMM1 + BN statistics
    k_gemm1<<<N_NODES / 16, 128, 0, stream>>>(agg, W1, b1, h1, colsum, colsumsq);
    // 4) fold stats into scale/shift
    k_bn_finalize<<<1, 256, 0, stream>>>(colsum, colsumsq, gamma, beta, scale, shift);
    // 5) BN+ReLU fused GEMM2 -> out
    k_gemm2<<<N_NODES / 16, 128, 0, stream>>>(h1, scale, shift, W2, b2, out);
}